// ContinuousThoughtMachineSIMPLE_20607253086665
// MI455X (gfx1250) — compile-verified
//
#include <hip/hip_runtime.h>
#include <hip/hip_bf16.h>
#include <math.h>

typedef float v2f __attribute__((ext_vector_type(2)));
typedef float v8f __attribute__((ext_vector_type(8)));

#define B_    32
#define C_    512
#define S_    256
#define ITER_ 25
#define DM    2048
#define DI    512
#define HEADS_ 8
#define HD    64
#define NS    512
#define MEM_  25
#define MH_   64
#define OD    1000

__device__ __forceinline__ float sigm(float x) { return 1.f / (1.f + expf(-x)); }

// ---------------------------------------------------------------------------
// Generic fp32 GEMM: C[m,n] = sum_k A[m,k]*B[k,n] (+ bias[n]), strided.
// One wave32 computes a 32x16 output block (two 16x16 tiles sharing the B
// fragment) with V_WMMA_F32_16X16X4_F32.  Requirements: M % 32 == 0,
// K % 4 == 0.  N may be ragged: out-of-range columns load a clamped address
// (col 0) and are masked at store -- no divergence in the inner loop, EXEC
// stays all-ones for WMMA.
//   AK1: A k-stride == 1 (vector 64b loads)   BK1: B k-stride == 1
// ---------------------------------------------------------------------------
template<bool AK1, bool BK1>
__global__ void gemm16_f32(const float* __restrict__ A, long aRow, long aCol, long aBat,
                           const float* __restrict__ Bm, long bRow, long bCol,
                           const float* __restrict__ bias,
                           float* __restrict__ C, long cRow, long cCol, long cBat,
                           int M, int N, int K)
{
    A += (long)blockIdx.y * aBat;
    C += (long)blockIdx.y * cBat;
    int tilesN = (N + 15) >> 4;
    int tilesM2 = M >> 5;                          // pairs of 16-row tiles
    int wave = blockIdx.x * (blockDim.x >> 5) + (threadIdx.x >> 5);
    if (wave >= tilesM2 * tilesN) return;          // uniform per wave
    int tM = wave / tilesN, tN = wave % tilesN;
    int lane = threadIdx.x & 31;
    int r = lane & 15, h = lane >> 4;
    int bcol = tN * 16 + r;
    int bcolc = (bcol < N) ? bcol : 0;             // clamp; garbage masked at store

    const float* Ap0 = A + (long)(tM * 32 + r) * aRow;
    const float* Ap1 = Ap0 + 16 * aRow;
    const float* Bp  = Bm + (long)bcolc * bCol;

    v8f acc0 = {}, acc1 = {};
#pragma unroll 4
    for (int k0 = 0; k0 < K; k0 += 4) {
        int ka = k0 + 2 * h;
        v2f a0, a1, b;
        if (AK1) {
            a0 = *(const v2f*)(Ap0 + ka);
            a1 = *(const v2f*)(Ap1 + ka);
        } else {
            a0.x = Ap0[(long)ka * aCol]; a0.y = Ap0[(long)(ka + 1) * aCol];
            a1.x = Ap1[(long)ka * aCol]; a1.y = Ap1[(long)(ka + 1) * aCol];
        }
        if (BK1) {
            b = *(const v2f*)(Bp + ka);
        } else {
            b.x = Bp[(long)ka * bRow]; b.y = Bp[(long)(ka + 1) * bRow];
        }
        if (k0 + 64 < K)                           // stream weights toward WGP
            __builtin_prefetch(Bp + (long)(k0 + 64 + 2 * h) * bRow, 0, 3);
        acc0 = __builtin_amdgcn_wmma_f32_16x16x4_f32(false, a0, false, b,
                                                     (short)0, acc0, false, false);
        acc1 = __builtin_amdgcn_wmma_f32_16x16x4_f32(false, a1, false, b,
                                                     (short)0, acc1, false, false);
    }
#pragma unroll
    for (int i = 0; i < 8; ++i) {
        int rr0 = tM * 32 + i + 8 * h;             // M%32==0 -> always in range
        int cc = tN * 16 + r;
        if (cc < N) {
            float bv = bias ? bias[cc] : 0.f;
            C[(long)rr0 * cRow + (long)cc * cCol] = acc0[i] + bv;
            C[(long)(rr0 + 16) * cRow + (long)cc * cCol] = acc1[i] + bv;
        }
    }
}

// ----------------------- LayerNorm over dim 512 (in place) -----------------
__global__ void ln512_inplace(float* __restrict__ X, const float* __restrict__ g,
                              const float* __restrict__ b)
{
    __shared__ float red[256];
    int row = blockIdx.x, tid = threadIdx.x;
    float* x = X + (long)row * 512;
    float v0 = x[tid], v1 = x[tid + 256];
    red[tid] = v0 + v1; __syncthreads();
    for (int st = 128; st > 0; st >>= 1) { if (tid < st) red[tid] += red[tid + st]; __syncthreads(); }
    float mean = red[0] * (1.f / 512.f); __syncthreads();
    float d0 = v0 - mean, d1 = v1 - mean;
    red[tid] = d0 * d0 + d1 * d1; __syncthreads();
    for (int st = 128; st > 0; st >>= 1) { if (tid < st) red[tid] += red[tid + st]; __syncthreads(); }
    float inv = rsqrtf(red[0] * (1.f / 512.f) + 1e-5f);
    x[tid]       = d0 * inv * g[tid]       + b[tid];
    x[tid + 256] = d1 * inv * g[tid + 256] + b[tid + 256];
}

// -------- GLU(4096->2048) + LayerNorm, write state into trace ring slot ----
__global__ void glu_ln_ring(const float* __restrict__ syn, const float* __restrict__ g,
                            const float* __restrict__ bb, float* __restrict__ ring, int slot)
{
    __shared__ float red[256];
    int b = blockIdx.x, tid = threadIdx.x;
    const float* row = syn + (long)b * (2 * DM);
    float v[8];
    float s = 0.f;
#pragma unroll
    for (int k = 0; k < 8; ++k) {
        int j = tid + k * 256;
        v[k] = row[j] * sigm(row[j + DM]);
        s += v[k];
    }
    red[tid] = s; __syncthreads();
    for (int st = 128; st > 0; st >>= 1) { if (tid < st) red[tid] += red[tid + st]; __syncthreads(); }
    float mean = red[0] * (1.f / (float)DM); __syncthreads();
    float s2 = 0.f;
#pragma unroll
    for (int k = 0; k < 8; ++k) { float d = v[k] - mean; s2 += d * d; }
    red[tid] = s2; __syncthreads();
    for (int st = 128; st > 0; st >>= 1) { if (tid < st) red[tid] += red[tid + st]; __syncthreads(); }
    float inv = rsqrtf(red[0] * (1.f / (float)DM) + 1e-5f);
#pragma unroll
    for (int k = 0; k < 8; ++k) {
        int j = tid + k * 256;
        ring[((long)b * DM + j) * MEM_ + slot] = (v[k] - mean) * inv * g[j] + bb[j];
    }
}

// ----------------------- attention: one block per (b,h) --------------------
__global__ void attn_kernel(const float* __restrict__ q, const float* __restrict__ Kb,
                            const float* __restrict__ Vb, float* __restrict__ attn)
{
    __shared__ float sc[256];
    __shared__ float red[256];
    __shared__ float part[256];
    int bh = blockIdx.x; int b = bh >> 3; int hh = bh & 7;
    int tid = threadIdx.x;
    const float* qv = q + (long)b * DI + hh * HD;
    const float* Krow = Kb + ((long)(b * S_ + tid)) * DI + hh * HD;
    float dot = 0.f;
#pragma unroll 8
    for (int d = 0; d < HD; ++d) dot += qv[d] * Krow[d];
    sc[tid] = dot * 0.125f;
    red[tid] = sc[tid]; __syncthreads();
    for (int st = 128; st > 0; st >>= 1) { if (tid < st) red[tid] = fmaxf(red[tid], red[tid + st]); __syncthreads(); }
    float m = red[0]; __syncthreads();
    float e = expf(sc[tid] - m);
    sc[tid] = e;
    red[tid] = e; __syncthreads();
    for (int st = 128; st > 0; st >>= 1) { if (tid < st) red[tid] += red[tid + st]; __syncthreads(); }
    float sum = red[0]; __syncthreads();
    int d = tid & 63, qn = tid >> 6;
    const float* Vcol = Vb + ((long)(b * S_)) * DI + hh * HD + d;
    float acc = 0.f;
    for (int ss = qn * 64; ss < qn * 64 + 64; ++ss) acc += sc[ss] * Vcol[(long)ss * DI];
    part[tid] = acc; __syncthreads();
    if (tid < 64) {
        float o = (part[tid] + part[tid + 64] + part[tid + 128] + part[tid + 192]) / sum;
        attn[(long)b * DI + hh * HD + tid] = o;
    }
}

// -------- per-neuron NLM: trace(25) -> GLU(128->64) -> GLU(2->1) -----------
__global__ void nlm_kernel(const float* __restrict__ ring, int t,
                           const float* __restrict__ w1, const float* __restrict__ b1,
                           const float* __restrict__ w2, const float* __restrict__ b2,
                           float* __restrict__ act)
{
    int idx = blockIdx.x * blockDim.x + threadIdx.x;   // (b,d) pair, d fastest
    int d = idx & (DM - 1);
    float tr[MEM_];
    int start = (t + 1) % MEM_;
    long base = (long)idx * MEM_;
#pragma unroll
    for (int m = 0; m < MEM_; ++m) {
        int p = start + m; if (p >= MEM_) p -= MEM_;
        tr[m] = ring[base + p];
    }
    float a0 = b2[d * 2 + 0], a1 = b2[d * 2 + 1];
    for (int hh = 0; hh < MH_; ++hh) {
        float x0 = b1[(long)d * 128 + hh];
        float x1 = b1[(long)d * 128 + 64 + hh];
        const float* w1p0 = w1 + (long)hh * DM + d;
        const float* w1p1 = w1 + (long)(hh + 64) * DM + d;
#pragma unroll
        for (int m = 0; m < MEM_; ++m) {
            x0 += tr[m] * w1p0[(long)m * 128 * DM];
            x1 += tr[m] * w1p1[(long)m * 128 * DM];
        }
        float hv = x0 * sigm(x1);
        a0 += hv * w2[(long)(hh * 2 + 0) * DM + d];
        a1 += hv * w2[(long)(hh * 2 + 1) * DM + d];
    }
    act[idx] = a0 * sigm(a1);
}

// ------------- synchronisation update (shared by action/out paths) ---------
__global__ void sync_kernel(const float* __restrict__ act, const int* __restrict__ il,
                            const int* __restrict__ ir, const float* __restrict__ decay,
                            float* __restrict__ da, float* __restrict__ db,
                            float* __restrict__ syncv)
{
    int idx = blockIdx.x * blockDim.x + threadIdx.x;   // B*NS
    int b = idx >> 9, n = idx & 511;
    float r = expf(-fminf(fmaxf(decay[n], 0.f), 15.f));
    float v = r * da[idx] + act[(long)b * DM + il[n]] * act[(long)b * DM + ir[n]];
    float w = r * db[idx] + 1.f;
    da[idx] = v; db[idx] = w;
    syncv[idx] = v / sqrtf(w);
}

__global__ void init_state_kernel(const float* __restrict__ s0, const float* __restrict__ tr0,
                                  float* __restrict__ act, float* __restrict__ ring)
{
    int idx = blockIdx.x * blockDim.x + threadIdx.x;   // B*DM
    int d = idx & (DM - 1);
    act[idx] = s0[d];
    long base = (long)idx * MEM_;
#pragma unroll
    for (int m = 0; m < MEM_; ++m) ring[base + m] = tr0[(long)d * MEM_ + m];
}

__global__ void init_sync_kernel(const float* __restrict__ s0, const int* __restrict__ iol,
                                 const int* __restrict__ ior,
                                 float* __restrict__ daa, float* __restrict__ dba,
                                 float* __restrict__ dao, float* __restrict__ dbo)
{
    int idx = blockIdx.x * blockDim.x + threadIdx.x;   // B*NS
    int n = idx & 511;
    daa[idx] = 0.f; dba[idx] = 0.f;
    dao[idx] = s0[iol[n]] * s0[ior[n]];
    dbo[idx] = 1.f;
}

__global__ void concat_kernel(const float* __restrict__ a, const float* __restrict__ b,
                              float* __restrict__ pre)
{
    int idx = blockIdx.x * blockDim.x + threadIdx.x;   // B*(DI+DM)
    int bb = idx / (DI + DM); int j = idx - bb * (DI + DM);
    pre[idx] = (j < DI) ? a[(long)bb * DI + j] : b[(long)bb * DM + (j - DI)];
}

__global__ void entropy_kernel(float* __restrict__ out, int t)
{
    __shared__ float red[256];
    int b = blockIdx.x, tid = threadIdx.x;
    const float* pr = out + (long)b * (OD * ITER_) + t;   // stride ITER_
    float mx = -INFINITY;
    for (int o = tid; o < OD; o += 256) mx = fmaxf(mx, pr[(long)o * ITER_]);
    red[tid] = mx; __syncthreads();
    for (int st = 128; st > 0; st >>= 1) { if (tid < st) red[tid] = fmaxf(red[tid], red[tid + st]); __syncthreads(); }
    float m = red[0]; __syncthreads();
    float se = 0.f;
    for (int o = tid; o < OD; o += 256) se += expf(pr[(long)o * ITER_] - m);
    red[tid] = se; __syncthreads();
    for (int st = 128; st > 0; st >>= 1) { if (tid < st) red[tid] += red[tid + st]; __syncthreads(); }
    float logZ = m + logf(red[0]); __syncthreads();
    float s2 = 0.f;
    for (int o = tid; o < OD; o += 256) { float lp = pr[(long)o * ITER_] - logZ; s2 += expf(lp) * lp; }
    red[tid] = s2; __syncthreads();
    for (int st = 128; st > 0; st >>= 1) { if (tid < st) red[tid] += red[tid + st]; __syncthreads(); }
    if (tid == 0) {
        float ne = -red[0] / logf((float)OD);
        float* cert = out + (long)B_ * OD * ITER_;
        cert[(long)b * 2 * ITER_ + t] = ne;
        cert[(long)b * 2 * ITER_ + ITER_ + t] = 1.f - ne;
    }
}

__global__ void copy_sync_kernel(const float* __restrict__ syO, float* __restrict__ out)
{
    int idx = blockIdx.x * blockDim.x + threadIdx.x;   // B*NS
    out[(long)B_ * OD * ITER_ + (long)B_ * 2 * ITER_ + idx] = syO[idx];
}

// ---------------------------------------------------------------------------
static void launch_gemm(hipStream_t s, bool ak1, bool bk1,
                        const float* A, long aR, long aC, long aB,
                        const float* Bm, long bR, long bC, const float* bias,
                        float* C, long cR, long cC, long cB,
                        int M, int N, int K, int batch)
{
    int tiles = (M / 32) * ((N + 15) / 16);
    dim3 grid((tiles + 3) / 4, batch);
    if (ak1 && bk1)
        gemm16_f32<true, true><<<grid, 128, 0, s>>>(A, aR, aC, aB, Bm, bR, bC, bias, C, cR, cC, cB, M, N, K);
    else if (ak1 && !bk1)
        gemm16_f32<true, false><<<grid, 128, 0, s>>>(A, aR, aC, aB, Bm, bR, bC, bias, C, cR, cC, cB, M, N, K);
    else if (!ak1 && bk1)
        gemm16_f32<false, true><<<grid, 128, 0, s>>>(A, aR, aC, aB, Bm, bR, bC, bias, C, cR, cC, cB, M, N, K);
    else
        gemm16_f32<false, false><<<grid, 128, 0, s>>>(A, aR, aC, aB, Bm, bR, bC, bias, C, cR, cC, cB, M, N, K);
}

extern "C" void kernel_launch(void* const* d_in, const int* in_sizes, int n_in,
                              void* d_out, int out_size, void* d_ws, size_t ws_size,
                              hipStream_t stream)
{
    (void)in_sizes; (void)n_in; (void)out_size; (void)ws_size;
    const float* x          = (const float*)d_in[0];
    const float* kv_w       = (const float*)d_in[1];
    const float* kv_b       = (const float*)d_in[2];
    const float* kv_ln_g    = (const float*)d_in[3];
    const float* kv_ln_b    = (const float*)d_in[4];
    const float* q_w        = (const float*)d_in[5];
    const float* q_b        = (const float*)d_in[6];
    const float* attn_in_w  = (const float*)d_in[7];
    const float* attn_in_b  = (const float*)d_in[8];
    const float* attn_out_w = (const float*)d_in[9];
    const float* attn_out_b = (const float*)d_in[10];
    const float* syn_w      = (const float*)d_in[11];
    const float* syn_b      = (const float*)d_in[12];
    const float* syn_ln_g   = (const float*)d_in[13];
    const float* syn_ln_b   = (const float*)d_in[14];
    const float* tp1_w      = (const float*)d_in[15];
    const float* tp1_b      = (const float*)d_in[16];
    const float* tp2_w      = (const float*)d_in[17];
    const float* tp2_b      = (const float*)d_in[18];
    const float* s0         = (const float*)d_in[19];
    const float* tr0        = (const float*)d_in[20];
    const float* dec_a      = (const float*)d_in[21];
    const float* dec_o      = (const float*)d_in[22];
    const float* out_w      = (const float*)d_in[23];
    const float* out_b      = (const float*)d_in[24];
    const int*   ial        = (const int*)d_in[25];
    const int*   iar        = (const int*)d_in[26];
    const int*   iol        = (const int*)d_in[27];
    const int*   ior        = (const int*)d_in[28];
    float* out = (float*)d_out;

    // workspace carve (floats)
    float* W    = (float*)d_ws;
    float* kv   = W; W += (long)B_ * S_ * DI;
    float* Kb   = W; W += (long)B_ * S_ * DI;
    float* Vb   = W; W += (long)B_ * S_ * DI;
    float* ring = W; W += (long)B_ * DM * MEM_;
    float* act  = W; W += (long)B_ * DM;
    float* syn  = W; W += (long)B_ * 2 * DM;
    float* pre  = W; W += (long)B_ * (DI + DM);
    float* attnb= W; W += (long)B_ * DI;
    float* attno= W; W += (long)B_ * DI;
    float* qtmp = W; W += (long)B_ * DI;
    float* qb   = W; W += (long)B_ * DI;
    float* syA  = W; W += (long)B_ * NS;
    float* syO  = W; W += (long)B_ * NS;
    float* daa  = W; W += (long)B_ * NS;
    float* dba  = W; W += (long)B_ * NS;
    float* dao  = W; W += (long)B_ * NS;
    float* dbo  = W; W += (long)B_ * NS;

    // ---- precompute: kv = LN(x^T @ kv_w + kv_b); K,V projections ----------
    // A[row=s][k=c] = x[b,c,s]  (aRow=1, aCol=S_)
    launch_gemm(stream, false, false,
                x, 1, S_, (long)C_ * S_,
                kv_w, DI, 1, kv_b,
                kv, DI, 1, (long)S_ * DI,
                S_, DI, C_, B_);
    ln512_inplace<<<B_ * S_, 256, 0, stream>>>(kv, kv_ln_g, kv_ln_b);
    // K = kv @ wk^T + bk : B[k=d][n=e] = attn_in_w[(DI+e)*DI + d]  (bRow=1)
    launch_gemm(stream, true, true,
                kv, DI, 1, (long)S_ * DI,
                attn_in_w + (long)DI * DI, 1, DI, attn_in_b + DI,
                Kb, DI, 1, (long)S_ * DI,
                S_, DI, DI, B_);
    launch_gemm(stream, true, true,
                kv, DI, 1, (long)S_ * DI,
                attn_in_w + 2l * DI * DI, 1, DI, attn_in_b + 2 * DI,
                Vb, DI, 1, (long)S_ * DI,
                S_, DI, DI, B_);

    init_state_kernel<<<(B_ * DM) / 256, 256, 0, stream>>>(s0, tr0, act, ring);
    init_sync_kernel<<<(B_ * NS) / 256, 256, 0, stream>>>(s0, iol, ior, daa, dba, dao, dbo);

    for (int t = 0; t < ITER_; ++t) {
        // action synchronisation -> syA
        sync_kernel<<<(B_ * NS) / 256, 256, 0, stream>>>(act, ial, iar, dec_a, daa, dba, syA);
        // q = (syA @ q_w + q_b) @ wq^T + bq
        launch_gemm(stream, true, false,
                    syA, NS, 1, 0, q_w, DI, 1, q_b,
                    qtmp, DI, 1, 0, B_, DI, NS, 1);
        launch_gemm(stream, true, true,
                    qtmp, DI, 1, 0, attn_in_w, 1, DI, attn_in_b,
                    qb, DI, 1, 0, B_, DI, DI, 1);
        // attention
        attn_kernel<<<B_ * HEADS_, 256, 0, stream>>>(qb, Kb, Vb, attnb);
        // attn @ attn_out_w^T + attn_out_b
        launch_gemm(stream, true, true,
                    attnb, DI, 1, 0, attn_out_w, 1, DI, attn_out_b,
                    attno, DI, 1, 0, B_, DI, DI, 1);
        // pre = concat(attn, act)
        concat_kernel<<<(B_ * (DI + DM)) / 256, 256, 0, stream>>>(attno, act, pre);
        // syn = pre @ syn_w + syn_b   (dominant GEMM: 32x4096x2560)
        launch_gemm(stream, true, false,
                    pre, DI + DM, 1, 0, syn_w, 2 * DM, 1, syn_b,
                    syn, 2 * DM, 1, 0, B_, 2 * DM, DI + DM, 1);
        // state = LN(GLU(syn)) -> trace ring slot
        glu_ln_ring<<<B_, 256, 0, stream>>>(syn, syn_ln_g, syn_ln_b, ring, t % MEM_);
        // per-neuron NLMs -> act
        nlm_kernel<<<(B_ * DM) / 256, 256, 0, stream>>>(ring, t, tp1_w, tp1_b, tp2_w, tp2_b, act);
        // out synchronisation -> syO
        sync_kernel<<<(B_ * NS) / 256, 256, 0, stream>>>(act, iol, ior, dec_o, dao, dbo, syO);
        // pred -> d_out (B, OD, ITER): row stride OD*ITER_, col stride ITER_, offset t
        launch_gemm(stream, true, false,
                    syO, NS, 1, 0, out_w, OD, 1, out_b,
                    out + t, (long)OD * ITER_, ITER_, 0, B_, OD, NS, 1);
        // certainties
        entropy_kernel<<<B_, 256, 0, stream>>>(out, t);
    }
    // final sync_o output
    copy_sync_kernel<<<(B_ * NS) / 256, 256, 0, stream>>>(syO, out);
}